// Custom_LSTM_91190745628723
// MI455X (gfx1250) — compile-verified
//
#include <hip/hip_runtime.h>
#include <math.h>

#define H 4096
#define K2 (2 * H)                    // 8192 columns per weight row
#define ROWS 16                       // rows per workgroup (one WMMA M-tile)
#define WAVES 8
#define THREADS 256
#define K_PER_WAVE (K2 / WAVES)       // 1024
#define CHUNKS (K_PER_WAVE / 128)     // 8 float4 chunks per lane

typedef float vf4 __attribute__((ext_vector_type(4)));
typedef float v2f __attribute__((ext_vector_type(2)));
typedef float v8f __attribute__((ext_vector_type(8)));

__device__ __forceinline__ void wait_dscnt0() {
#if __has_builtin(__builtin_amdgcn_s_wait_dscnt)
    __builtin_amdgcn_s_wait_dscnt(0);
    asm volatile("" ::: "memory");
#else
    asm volatile("s_wait_dscnt 0" ::: "memory");
#endif
}

// Pass 1: 4 gates x 256 row-blocks = 1024 workgroups. Each workgroup produces
// 16 bias-added pre-activations for one gate. Weights streamed exactly once
// with non-temporal b128 loads; z fragment cached in VGPRs.
__global__ __launch_bounds__(THREADS)
void lstm_gate_gemv(const float* __restrict__ h,
                    const float* __restrict__ c,
                    const float* __restrict__ x,
                    const float* __restrict__ Wf, const float* __restrict__ bf,
                    const float* __restrict__ Wi, const float* __restrict__ bi,
                    const float* __restrict__ WC, const float* __restrict__ bC,
                    const float* __restrict__ Wo, const float* __restrict__ bo,
                    float* __restrict__ preact)
{
    __shared__ float red[WAVES * 64];   // per-wave 4 group-sums x 16 rows
    __shared__ float outp[WAVES * 16];  // per-wave 16 row results

    const int tid  = threadIdx.x;
    const int lane = tid & 31;
    const int wave = tid >> 5;

    const int gate = blockIdx.x >> 8;         // 0..3
    const int m0   = (blockIdx.x & 255) * ROWS;

    const float* W;
    const float* b;
    switch (gate) {
        case 0:  W = Wf; b = bf; break;
        case 1:  W = Wi; b = bi; break;
        case 2:  W = WC; b = bC; break;       // candidate gate: z = [h, c] quirk
        default: W = Wo; b = bo; break;
    }

    // This wave's K-slice [k0, k0+1024) lies entirely in one half of z.
    const int k0 = wave * K_PER_WAVE;
    const float* zsrc;
    int zoff;
    if (k0 < H) { zsrc = h; zoff = k0; }
    else        { zsrc = (gate == 2) ? c : x; zoff = k0 - H; }

    // Preload this lane's 32 z values (reused across all 16 rows).
    vf4 zr[CHUNKS];
#pragma unroll
    for (int i = 0; i < CHUNKS; ++i)
        zr[i] = *(const vf4*)(zsrc + zoff + i * 128 + lane * 4);

    float acc[ROWS];

    // Hot loop: for each row, 8 coalesced NT b128 loads + 32 FMAs per lane.
    // Wave reads 512 contiguous bytes per load instruction.
#pragma unroll
    for (int r = 0; r < ROWS; ++r) {
        const float* p = W + (size_t)(m0 + r) * K2 + k0 + lane * 4;
        float a = 0.f;
#pragma unroll
        for (int i = 0; i < CHUNKS; ++i) {
            vf4 w4 = __builtin_nontemporal_load((const vf4*)(p + i * 128));
            a = fmaf(w4.x, zr[i].x, a);
            a = fmaf(w4.y, zr[i].y, a);
            a = fmaf(w4.z, zr[i].z, a);
            a = fmaf(w4.w, zr[i].w, a);
        }
        acc[r] = a;
    }

    // Intra-8-lane tree: every lane ends holding its 8-lane group sum g_j[r],
    // where j = lane>>3 selects one of 4 groups.
#pragma unroll
    for (int r = 0; r < ROWS; ++r) {
        acc[r] += __shfl_xor(acc[r], 1, 32);
        acc[r] += __shfl_xor(acc[r], 2, 32);
        acc[r] += __shfl_xor(acc[r], 4, 32);
    }

    // Stage group sums to LDS in V_WMMA A-matrix (16x4 f32) order.
    float* wred = &red[wave * 64];
    if ((lane & 7) == 0) {
        const int g = lane >> 3;
#pragma unroll
        for (int r = 0; r < ROWS; ++r)
            wred[g * 16 + r] = acc[r];
    }
    wait_dscnt0();

    // A[m,j]: lanes 0-15 hold (K=0,K=1) for row m=lane; lanes 16-31 hold
    // (K=2,K=3) for row m=lane-16 (ISA 7.12.2, 32-bit A 16x4 layout).
    v2f A;
    if (lane < 16) { A.x = wred[lane];            A.y = wred[16 + lane];        }
    else           { A.x = wred[32 + (lane - 16)]; A.y = wred[48 + (lane - 16)]; }

    // B = all-ones 4x16 (layout-independent); D[m,n] = sum_j A[m,j] = row sum.
    v2f B; B.x = 1.0f; B.y = 1.0f;
    v8f Cz = {};
    v8f D = __builtin_amdgcn_wmma_f32_16x16x4_f32(
        /*neg_a=*/false, A, /*neg_b=*/false, B,
        /*c_mod=*/(short)0, Cz, /*reuse_a=*/false, /*reuse_b=*/false);

    // Column N=0 of D: lane 0 carries rows 0..7 in D[0..7], lane 16 rows 8..15.
    if (lane == 0 || lane == 16) {
        const int base = (lane == 0) ? 0 : 8;
#pragma unroll
        for (int i = 0; i < 8; ++i)
            outp[wave * 16 + base + i] = D[i];
    }
    __syncthreads();

    // Combine the 8 wave K-slices, add bias, emit pre-activation.
    if (tid < ROWS) {
        float s = 0.f;
#pragma unroll
        for (int w = 0; w < WAVES; ++w) s += outp[w * 16 + tid];
        preact[gate * H + m0 + tid] = s + b[m0 + tid];
    }
}

// Pass 2: pointwise gate math. Negligible cost (H = 4096 elements).
__global__ __launch_bounds__(THREADS)
void lstm_pointwise(const float* __restrict__ pre,
                    const float* __restrict__ c_prev,
                    float* __restrict__ out)
{
    const int t = blockIdx.x * blockDim.x + threadIdx.x;
    if (t >= H) return;
    const float pf = pre[t];
    const float pi = pre[H + t];
    const float pc = pre[2 * H + t];
    const float po = pre[3 * H + t];

    const float f  = 1.f / (1.f + expf(-pf));
    const float i  = 1.f / (1.f + expf(-pi));
    const float ct = tanhf(pc);
    const float o  = 1.f / (1.f + expf(-po));

    const float C = f * c_prev[t] + i * ct;
    const float hh = o * tanhf(C);

    out[t]     = hh;  // h_t
    out[H + t] = C;   // C_t
}

extern "C" void kernel_launch(void* const* d_in, const int* in_sizes, int n_in,
                              void* d_out, int out_size, void* d_ws, size_t ws_size,
                              hipStream_t stream) {
    (void)in_sizes; (void)n_in; (void)out_size; (void)ws_size;
    const float* h  = (const float*)d_in[0];
    const float* c  = (const float*)d_in[1];
    const float* x  = (const float*)d_in[2];
    const float* Wf = (const float*)d_in[3];
    const float* bf = (const float*)d_in[4];
    const float* Wi = (const float*)d_in[5];
    const float* bi = (const float*)d_in[6];
    const float* WC = (const float*)d_in[7];
    const float* bC = (const float*)d_in[8];
    const float* Wo = (const float*)d_in[9];
    const float* bo = (const float*)d_in[10];

    float* preact = (float*)d_ws;           // 4*H floats = 64 KB scratch
    float* out    = (float*)d_out;          // [h(4096), C(4096)]

    lstm_gate_gemv<<<dim3(4 * (H / ROWS)), dim3(THREADS), 0, stream>>>(
        h, c, x, Wf, bf, Wi, bi, WC, bC, Wo, bo, preact);
    lstm_pointwise<<<dim3(H / THREADS), dim3(THREADS), 0, stream>>>(preact, c, out);
}